// GraphAttentionLayer_120259085136
// MI455X (gfx1250) — compile-verified
//
#include <hip/hip_runtime.h>
#include <hip/hip_bf16.h>

#define ALPHA   0.2f
#define NEG_INF -9000000000000000.0f
#define BG  96      // B*G = 8*12
#define V_  512
#define IN_ 64
#define OUT_ 64
#define AP  516     // kernel-3 LDS att row pitch (bank-conflict-free vs 512)
#define HP  68      // kernel-1 LDS pitch: 16B-aligned rows, lanes spread 4 banks apart

typedef float v2f __attribute__((ext_vector_type(2)));
typedef float v8f __attribute__((ext_vector_type(8)));

__device__ __forceinline__ float lrelu(float x) { return x >= 0.0f ? x : ALPHA * x; }

// Raw LDS byte offset of a __shared__ object (generic -> addrspace(3) cast).
__device__ __forceinline__ unsigned lds_offset(const void* p) {
  return (unsigned)(size_t)(__attribute__((address_space(3))) const void*)p;
}

// ---------------------------------------------------------------------------
// Kernel 1: Wh[bg, v, :] = lrelu( h[bg, v, :] @ W[v] )
// One block per node v. Stage h-slice (96x64) + W_v^T (64x64) into LDS with
// coalesced b128/b32 global loads, then 24 waves run 16 f32 WMMAs each from
// conflict-free LDS b64 reads (fused to ds_load_2addr_b64 by the backend).
// ---------------------------------------------------------------------------
__global__ __launch_bounds__(768) void gat_wh_kernel(
    const float* __restrict__ h, const float* __restrict__ W,
    float* __restrict__ Wh)
{
  __shared__ float hs[BG * HP];     // hs[bg][i]  (pitch 68)
  __shared__ float wt[OUT_ * HP];   // wt[o][i]   (W_v transposed, pitch 68)

  const int v = blockIdx.x;
  const int t = threadIdx.x;

  // ---- stage h[0..95][v][0..63]: 1536 float4s, 2 per thread, coalesced ----
#pragma unroll
  for (int j = 0; j < 2; ++j) {
    int fidx = t + j * 768;
    int row  = fidx >> 4;             // 16 float4 per 64-float row
    int c4   = (fidx & 15) << 2;
    float4 val = *(const float4*)(h + ((size_t)row * V_ + v) * IN_ + c4);
    *(float4*)&hs[row * HP + c4] = val;   // (row*68 + c4)*4 is 16B aligned
  }
  // ---- stage W_v transposed: wt[o][i] = W[v][i][o], coalesced global reads ----
  const float* Wv = W + (size_t)v * IN_ * OUT_;
#pragma unroll
  for (int idx = t; idx < IN_ * OUT_; idx += 768) {
    int i = idx >> 6;                 // input dim (row of W_v)
    int o = idx & 63;                 // output dim (col of W_v)
    wt[o * HP + i] = Wv[idx];
  }
  __syncthreads();

  const int wid    = t >> 5;
  const int lane   = t & 31;
  const int mtile  = wid >> 2;
  const int ntile  = wid & 3;
  const int lane16 = lane & 15;
  const int khalf  = (lane >> 4) << 1;            // K = vgpr + khalf

  const int bgA  = mtile * 16 + lane16;           // A-frag row (M)
  const int colN = ntile * 16 + lane16;           // B-frag / D column (N)

  v8f acc = {0.f, 0.f, 0.f, 0.f, 0.f, 0.f, 0.f, 0.f};
#pragma unroll
  for (int k = 0; k < IN_; k += 4) {
    v2f av = *(const v2f*)&hs[bgA  * HP + k + khalf];  // A[M][K], A[M][K+1]
    v2f bv = *(const v2f*)&wt[colN * HP + k + khalf];  // B[K][N], B[K+1][N]
    acc = __builtin_amdgcn_wmma_f32_16x16x4_f32(
        false, av, false, bv, (short)0, acc, false, false);
  }

  const int moff = (lane >> 4) * 8;               // D: vgpr r -> M = r + 8*lanehalf
#pragma unroll
  for (int r = 0; r < 8; ++r) {
    int bgD = mtile * 16 + r + moff;
    Wh[((size_t)bgD * V_ + v) * OUT_ + colN] = lrelu(acc[r]);   // coalesced b32
  }
}

// ---------------------------------------------------------------------------
// Kernel 2: Wh1[bg,v] = Wh[bg,v,:].a1[v],  Wh2[bg,v] = Wh[bg,v,:].a2[v]
// One wave per (bg, v) pair; wave32 shuffle reduction.
// ---------------------------------------------------------------------------
__global__ __launch_bounds__(256) void gat_dots_kernel(
    const float* __restrict__ Wh, const float* __restrict__ a,
    float* __restrict__ Wh1, float* __restrict__ Wh2)
{
  const int pair = (int)(blockIdx.x * (blockDim.x >> 5)) + (threadIdx.x >> 5);
  const int lane = threadIdx.x & 31;
  if (pair >= BG * V_) return;
  const int bg = pair / V_;
  const int v  = pair % V_;
  const float* row = Wh + ((size_t)bg * V_ + v) * OUT_;
  const float* av  = a + (size_t)v * 2 * OUT_;

  float s1 = row[lane] * av[lane]        + row[lane + 32] * av[lane + 32];
  float s2 = row[lane] * av[OUT_ + lane] + row[lane + 32] * av[OUT_ + lane + 32];
#pragma unroll
  for (int off = 16; off > 0; off >>= 1) {
    s1 += __shfl_xor(s1, off, 32);
    s2 += __shfl_xor(s2, off, 32);
  }
  if (lane == 0) { Wh1[pair] = s1; Wh2[pair] = s2; }
}

// ---------------------------------------------------------------------------
// Kernel 3: masked softmax + h_prime = att @ Wh_bg, lrelu epilogue.
// Block = (vtile, bg): 16 attention rows x 512 cols.  4 waves (ntile 0..3).
// Phase A: exp(e - rowmax) into LDS (unnormalized), invsum per row.
// Phase B: 16x512 @ 512x64.  Each wave DMAs its 64x16 B-slab of Wh_bg into a
//          private LDS double buffer with GLOBAL_LOAD_ASYNC_TO_LDS_B128
//          (ASYNCcnt-tracked), overlapping DMA of chunk c+1 with the 16 f32
//          WMMAs consuming chunk c.  Scale by invsum; lrelu; store.
// ---------------------------------------------------------------------------
__global__ __launch_bounds__(128) void gat_attn_kernel(
    const int* __restrict__ adj, const float* __restrict__ Wh,
    const float* __restrict__ Wh1, const float* __restrict__ Wh2,
    float* __restrict__ out)
{
  __shared__ float att[16 * AP];
  __shared__ float invsum[16];
  __shared__ float ldsB[4 * 2 * 64 * 16];   // [wave][slot][64 rows][16 cols] = 32KB

  const int vtile = blockIdx.x;   // 0..31
  const int bg    = blockIdx.y;   // 0..95
  const int t     = threadIdx.x;

  // ---- Phase A: masked e, row softmax (8 lanes per row) ----
  {
    const int r = t >> 3;                         // 0..15
    const int s = t & 7;                          // 0..7
    const int v = vtile * 16 + r;
    const float w1 = Wh1[bg * V_ + v];
    const float* w2row  = Wh2 + (size_t)bg * V_;
    const int*   adjrow = adj + (size_t)v * V_;

    float mx = NEG_INF;
    for (int u = s; u < V_; u += 8) {
      float e = lrelu(w1 + w2row[u]);
      float m = (adjrow[u] > 0) ? e : NEG_INF;
      att[r * AP + u] = m;
      mx = fmaxf(mx, m);
    }
#pragma unroll
    for (int off = 1; off < 8; off <<= 1) mx = fmaxf(mx, __shfl_xor(mx, off, 32));

    float sum = 0.f;
    for (int u = s; u < V_; u += 8) {
      float p = __expf(att[r * AP + u] - mx);
      att[r * AP + u] = p;
      sum += p;
    }
#pragma unroll
    for (int off = 1; off < 8; off <<= 1) sum += __shfl_xor(sum, off, 32);
    if (s == 0) invsum[r] = 1.0f / sum;
  }
  __syncthreads();

  // ---- Phase B: attention matmul with async double-buffered B slabs ----
  const int wv     = t >> 5;                      // ntile 0..3
  const int lane   = t & 31;
  const int lane16 = lane & 15;
  const int khalf  = (lane >> 4) << 1;
  const int colN   = wv * 16 + lane16;
  const float* Whbg = Wh + (size_t)bg * V_ * OUT_;

  // Per-lane DMA coordinates: 8 rows x 16 cols per b128 op (32 lanes x 16B).
  const int r4 = lane >> 2;                       // row within 8-row group
  const int q4 = (lane & 3) << 2;                 // 4-float column quarter

  // Issue chunk 0 into slot 0.
  {
    unsigned lb = lds_offset(&ldsB[(wv * 2 + 0) * 1024 + r4 * 16 + q4]);
    const float* gb = Whbg + (size_t)r4 * OUT_ + wv * 16 + q4;
#pragma unroll
    for (int j = 0; j < 8; ++j) {
      asm volatile("global_load_async_to_lds_b128 %0, %1, off"
                   :: "v"(lb + j * 512),
                      "v"((unsigned long long)(size_t)(gb + j * 8 * OUT_))
                   : "memory");
    }
  }

  v8f acc = {0.f, 0.f, 0.f, 0.f, 0.f, 0.f, 0.f, 0.f};
  for (int c = 0; c < 8; ++c) {                   // 8 chunks of 64 K-rows
    if (c < 7) {                                  // prefetch chunk c+1
      int ss = (c + 1) & 1;
      unsigned lb = lds_offset(&ldsB[(wv * 2 + ss) * 1024 + r4 * 16 + q4]);
      const float* gb = Whbg + (size_t)((c + 1) * 64 + r4) * OUT_ + wv * 16 + q4;
#pragma unroll
      for (int j = 0; j < 8; ++j) {
        asm volatile("global_load_async_to_lds_b128 %0, %1, off"
                     :: "v"(lb + j * 512),
                        "v"((unsigned long long)(size_t)(gb + j * 8 * OUT_))
                     : "memory");
      }
      asm volatile("s_wait_asynccnt 0x8" ::: "memory");   // chunk c landed
    } else {
      asm volatile("s_wait_asynccnt 0x0" ::: "memory");   // last chunk landed
    }

    const float* arow = &att[lane16 * AP + c * 64];
    const float* brow = &ldsB[(wv * 2 + (c & 1)) * 1024];
#pragma unroll
    for (int kk = 0; kk < 64; kk += 4) {
      v2f av = *(const v2f*)(arow + kk + khalf);         // LDS b64, 8B aligned
      v2f bv;
      bv.x = brow[(kk + khalf + 0) * 16 + lane16];       // conflict-free b32
      bv.y = brow[(kk + khalf + 1) * 16 + lane16];
      acc = __builtin_amdgcn_wmma_f32_16x16x4_f32(
          false, av, false, bv, (short)0, acc, false, false);
    }
  }

  const int moff = (lane >> 4) * 8;
#pragma unroll
  for (int r = 0; r < 8; ++r) {
    int m = r + moff;                             // local row 0..15
    int v = vtile * 16 + m;
    float val = acc[r] * invsum[m];               // normalize, then lrelu
    out[((size_t)bg * V_ + v) * OUT_ + colN] = lrelu(val);
  }
}

// ---------------------------------------------------------------------------
extern "C" void kernel_launch(void* const* d_in, const int* in_sizes, int n_in,
                              void* d_out, int out_size, void* d_ws, size_t ws_size,
                              hipStream_t stream) {
  const float* h   = (const float*)d_in[0];   // (8,12,512,64) f32
  const int*   adj = (const int*)  d_in[1];   // (512,512) i32
  const float* W   = (const float*)d_in[2];   // (512,64,64) f32
  const float* a   = (const float*)d_in[3];   // (512,128,1) f32
  float* out = (float*)d_out;                 // (8,12,512,64) f32

  float* Wh  = (float*)d_ws;                            // BG*V*OUT f32 (12.6 MB)
  float* Wh1 = Wh  + (size_t)BG * V_ * OUT_;            // BG*V
  float* Wh2 = Wh1 + (size_t)BG * V_;                   // BG*V

  gat_wh_kernel  <<<V_, 768, 0, stream>>>(h, W, Wh);
  gat_dots_kernel<<<(BG * V_) / 8, 256, 0, stream>>>(Wh, a, Wh1, Wh2);
  gat_attn_kernel<<<dim3(32, BG), 128, 0, stream>>>(adj, Wh, Wh1, Wh2, out);
}